// TriPlaneGenerator_82240033784353
// MI455X (gfx1250) — compile-verified
//
#include <hip/hip_runtime.h>

// ---------------------------------------------------------------------------
// Tri-plane sampling for MI455X (gfx1250, wave32).
//   plane_features: [1, 3, 32, 256, 256] f32   (d_in[0])
//   coordinates:    [1, M, 3] f32              (d_in[1])
//   plane_axes:     [3, 3, 3] f32              (d_in[2])
//   box_warp:       python int scalar          (d_in[3], int32)
//   out:            [1, 3, M, 32] f32
//
// Strategy:
//   1) transpose features CHW -> HWC into d_ws so one wave32 (lane==channel)
//      reads each bilinear corner as ONE 128B cacheline from L2.
//   2) one wave handles 16 points of one plane; the 16x3 @ 3x2 projection is
//      done with a single V_WMMA_F32_16X16X4_F32 (K padded to 4, N cols 0..1).
//   3) broadcast projected coords with v_readlane (SGPR-uniform -> scalar
//      addressing for gathers), bilinear blend, NT store of the 128B output
//      vector (output stream = 201MB dominates HBM traffic).
// ---------------------------------------------------------------------------

typedef __attribute__((ext_vector_type(2))) float v2f;
typedef __attribute__((ext_vector_type(8))) float v8f;

#define TP_IH 256
#define TP_IW 256
#define TP_C  32
#define TP_P  3

// lane-constant broadcast without LDS (v_readlane_b32 -> SGPR)
__device__ __forceinline__ float tp_readlane_f(float v, int srcLane) {
  return __builtin_bit_cast(
      float, __builtin_amdgcn_readlane(__builtin_bit_cast(int, v), srcLane));
}

// ---------------- transpose [P,C,H,W] -> [P,H,W,C] --------------------------
__global__ __launch_bounds__(256) void tp_transpose_chw_hwc(
    const float* __restrict__ in, float* __restrict__ out) {
  __shared__ float tile[32][33];                 // +1 pad: no bank conflicts
  const int b    = blockIdx.x;                   // P*IH*(IW/32) blocks
  const int w0   = (b % (TP_IW / 32)) * 32;
  const int h    = (b / (TP_IW / 32)) % TP_IH;
  const int p    = b / ((TP_IW / 32) * TP_IH);
  const int lane = threadIdx.x & 31;
  const int grp  = threadIdx.x >> 5;             // 0..7
#pragma unroll
  for (int i = 0; i < 4; ++i) {                  // coalesced 128B reads
    const int c = grp * 4 + i;
    tile[c][lane] = in[(((long)(p * TP_C + c) * TP_IH + h) * TP_IW) + w0 + lane];
  }
  __syncthreads();
#pragma unroll
  for (int i = 0; i < 4; ++i) {                  // coalesced 128B writes
    const int w = grp * 4 + i;
    out[(((long)(p * TP_IH + h) * TP_IW) + w0 + w) * TP_C + lane] = tile[lane][w];
  }
}

// ---------------- projection (WMMA) + bilinear gather -----------------------
template <bool TRANSPOSED>
__global__ __launch_bounds__(256) void tp_sample(
    const float* __restrict__ feat,     // TRANSPOSED ? [P,IH,IW,C] : [P,C,IH,IW]
    const float* __restrict__ coords,   // [M,3]
    const float* __restrict__ axes,     // [P,3,3]
    const int*   __restrict__ boxwarp,  // [1]
    float*       __restrict__ out,      // [P,M,C]
    int M) {
  const int lane = threadIdx.x & 31;
  const int wave = threadIdx.x >> 5;
  const int groupsPerPlane = M >> 4;                       // 16 points / wave
  const long g = (long)blockIdx.x * 8 + wave;              // wave-task id
  if (g >= (long)TP_P * groupsPerPlane) return;            // wave-uniform exit
  const int p     = (int)(g / groupsPerPlane);
  const int mbase = (int)(g % groupsPerPlane) << 4;
  const float scale = 2.0f / (float)boxwarp[0];

  // ---- generic 3x3 inverse of plane p (adjugate / det; ~30 VALU ops) ----
  const float* ax = axes + p * 9;
  const float a00 = ax[0], a01 = ax[1], a02 = ax[2];
  const float a10 = ax[3], a11 = ax[4], a12 = ax[5];
  const float a20 = ax[6], a21 = ax[7], a22 = ax[8];
  const float det = a00 * (a11 * a22 - a12 * a21)
                  - a01 * (a10 * a22 - a12 * a20)
                  + a02 * (a10 * a21 - a11 * a20);
  const float rdet = 1.0f / det;
  // only columns d=0,1 of inv are consumed (proj keeps [..., :2])
  const float inv00 = (a11 * a22 - a12 * a21) * rdet;
  const float inv01 = (a02 * a21 - a01 * a22) * rdet;
  const float inv10 = (a12 * a20 - a10 * a22) * rdet;
  const float inv11 = (a00 * a22 - a02 * a20) * rdet;
  const float inv20 = (a10 * a21 - a11 * a20) * rdet;
  const float inv21 = (a01 * a20 - a00 * a21) * rdet;

  // ---- A matrix 16x4 f32 (ISA 7.12.2): lanes0-15 hold K0,K1; 16-31 K2,K3 ----
  const int mloc = lane & 15;
  const float* cp = coords + 3L * (mbase + mloc);
  v2f A;
  if (lane < 16) { A.x = cp[0] * scale; A.y = cp[1] * scale; }
  else           { A.x = cp[2] * scale; A.y = 0.0f; }

  // ---- B matrix 4x16 f32: B[k][n] = inv[k][n] for n<2, else 0 ----
  // layout mirrors A: vgpr0 = rows K0 (lanes0-15) / K2 (lanes16-31),
  //                   vgpr1 = rows K1            / K3(=0)
  const int n = lane & 15;
  v2f B;
  if (lane < 16) {
    B.x = (n == 0) ? inv00 : (n == 1) ? inv01 : 0.0f;   // K=0 row
    B.y = (n == 0) ? inv10 : (n == 1) ? inv11 : 0.0f;   // K=1 row
  } else {
    B.x = (n == 0) ? inv20 : (n == 1) ? inv21 : 0.0f;   // K=2 row
    B.y = 0.0f;                                         // K=3 row (pad)
  }

  // ---- D[m][d] = sum_k A[m][k] * B[k][d]  (projection for 16 points) ----
  v8f D = {};
  D = __builtin_amdgcn_wmma_f32_16x16x4_f32(
      /*neg_a=*/false, A, /*neg_b=*/false, B,
      /*c_mod=*/(short)0, D, /*reuse_a=*/false, /*reuse_b=*/false);

  const float* fp = feat + (long)p * TP_C * TP_IH * TP_IW;
  float* op = out + ((long)p * M + mbase) * TP_C + lane;

#pragma unroll
  for (int mm = 0; mm < 16; ++mm) {
    // D layout: VGPR r holds M=r (lanes0-15, N=lane) and M=r+8 (lanes16-31,
    // N=lane-16)  ->  point mm's (d0,d1) live in VGPR mm&7, lanes {0,1}/{16,17}
    const float de = D[mm & 7];
    const float px = tp_readlane_f(de, (mm < 8) ? 0 : 16);  // SGPR broadcast
    const float py = tp_readlane_f(de, (mm < 8) ? 1 : 17);  // (no LDS bpermute)

    const float u = (px + 1.0f) * 0.5f * (float)(TP_IW - 1);
    const float v = (py + 1.0f) * 0.5f * (float)(TP_IH - 1);
    const float x0f = floorf(u), y0f = floorf(v);
    // weights with UNclamped corners (matches reference)
    const float wx1 = u - x0f, wx0 = 1.0f - wx1;
    const float wy1 = v - y0f, wy0 = 1.0f - wy1;
    // clamped indices for the gathers
    const int x0 = min(max((int)x0f, 0), TP_IW - 1);
    const int y0 = min(max((int)y0f, 0), TP_IH - 1);
    const int x1 = min(max((int)x0f + 1, 0), TP_IW - 1);
    const int y1 = min(max((int)y0f + 1, 0), TP_IH - 1);

    float f00, f01, f10, f11;
    if (TRANSPOSED) {            // one 128B cacheline per corner per wave
      f00 = fp[((long)y0 * TP_IW + x0) * TP_C + lane];
      f01 = fp[((long)y0 * TP_IW + x1) * TP_C + lane];
      f10 = fp[((long)y1 * TP_IW + x0) * TP_C + lane];
      f11 = fp[((long)y1 * TP_IW + x1) * TP_C + lane];
    } else {                     // fallback: strided channel gather
      const float* fc = fp + (long)lane * TP_IH * TP_IW;
      f00 = fc[y0 * TP_IW + x0];
      f01 = fc[y0 * TP_IW + x1];
      f10 = fc[y1 * TP_IW + x0];
      f11 = fc[y1 * TP_IW + x1];
    }
    const float val = (f00 * wx0 + f01 * wx1) * wy0
                    + (f10 * wx0 + f11 * wx1) * wy1;
    // output is write-once streaming (201MB) -> non-temporal store
    __builtin_nontemporal_store(val, op + (long)mm * TP_C);
  }
}

// ---------------------------------------------------------------------------
extern "C" void kernel_launch(void* const* d_in, const int* in_sizes, int n_in,
                              void* d_out, int out_size, void* d_ws, size_t ws_size,
                              hipStream_t stream) {
  const float* feats   = (const float*)d_in[0];   // [1,3,32,256,256]
  const float* coords  = (const float*)d_in[1];   // [1,M,3]
  const float* axes    = (const float*)d_in[2];   // [3,3,3]
  const int*   boxwarp = (const int*)d_in[3];     // scalar int
  float*       out     = (float*)d_out;           // [1,3,M,32]
  const int M = in_sizes[1] / 3;

  const size_t featBytes = (size_t)TP_P * TP_IH * TP_IW * TP_C * sizeof(float);
  const long totalWaves  = (long)TP_P * (M >> 4);
  const int  sampleBlocks = (int)((totalWaves + 7) / 8);

  if (ws_size >= featBytes) {
    float* featT = (float*)d_ws;
    tp_transpose_chw_hwc<<<TP_P * TP_IH * (TP_IW / 32), 256, 0, stream>>>(feats, featT);
    tp_sample<true><<<sampleBlocks, 256, 0, stream>>>(featT, coords, axes, boxwarp, out, M);
  } else {
    tp_sample<false><<<sampleBlocks, 256, 0, stream>>>(feats, coords, axes, boxwarp, out, M);
  }
}